// MSEDSNTDistanceDoubleLoss_55035710931296
// MI455X (gfx1250) — compile-verified
//
#include <hip/hip_runtime.h>
#include <cstdint>

// B=32, C=2, H=W=512 -> 64 planes of 262144 floats.
// Single-sweep design: 64 blocks/plane (4096 blocks), 256 threads/block,
// 4096 elements per tensor per block = 4 stages of 1024 floats, staged
// into LDS once via async DMA; block-local softmax shift + online rescale
// in the combine kernel.

#define HW_CONST 262144
#define BPP      64          // blocks per plane
#define NBLK     4096        // 64 planes * BPP
#define EPB      4096        // elements per block per tensor

// ws layout (floats):
//  [0 .. 65536)  : NBLK rows of 16: {mi, mt, idx(int), Zi, Zt, Sx, Sy, Sii, Stt, Sit, pad..}
//  [65536 .. +512): 64 rows of 8 plane results {px, py, mse, ed, tx, ty, pad..}
enum : int { OFF_PART = 0, OFF_S3 = NBLK * 16 };

__device__ __forceinline__ void async_pair(unsigned li, const float* ai,
                                           unsigned lt, const float* at) {
  // GV-mode async DMA: per-lane LDS dest VGPR, 64-bit global addr VGPR pair.
  asm volatile(
      "global_load_async_to_lds_b128 %0, %1, off\n\t"
      "global_load_async_to_lds_b128 %2, %3, off"
      :: "v"(li), "v"((unsigned long long)(uintptr_t)ai),
         "v"(lt), "v"((unsigned long long)(uintptr_t)at)
      : "memory");
}

// -------------------------------------------------------------- main pass ---
__global__ __launch_bounds__(256) void k_main(const float* __restrict__ in,
                                              const float* __restrict__ tg,
                                              float* __restrict__ ws) {
  const int t     = threadIdx.x;
  const int blk   = blockIdx.x;          // plane*64 + blkin
  const int plane = blk >> 6;
  const int blkin = blk & 63;

  __shared__ float4 bufI[4][256];        // 16 KB input stage
  __shared__ float4 bufT[4][256];        // 16 KB target stage
  __shared__ float  smi[8], smt[8];
  __shared__ int    sit[8];
  __shared__ float  red[8][7];
  __shared__ float  bmax[2];
  __shared__ int    bidx[1];

  const size_t gbase = (size_t)plane * HW_CONST + (size_t)blkin * EPB;
  const float* gin = in + gbase + t * 4;
  const float* gtg = tg + gbase + t * 4;

  // Issue the block's entire 32 KB as 8 async DMA instructions per wave.
#pragma unroll
  for (int s = 0; s < 4; ++s) {
    async_pair((unsigned)(uintptr_t)&bufI[s][t], gin + s * 1024,
               (unsigned)(uintptr_t)&bufT[s][t], gtg + s * 1024);
  }

  // ---- Phase A: block-local max / argmax, overlapped with in-flight DMA ----
  float mi = -INFINITY, mt = -INFINITY;
  int   it = 0;
#pragma unroll
  for (int s = 0; s < 4; ++s) {
    switch (s) {  // asyncs complete in issue order: drain ladder
      case 0:  asm volatile("s_wait_asynccnt 6" ::: "memory"); break;
      case 1:  asm volatile("s_wait_asynccnt 4" ::: "memory"); break;
      case 2:  asm volatile("s_wait_asynccnt 2" ::: "memory"); break;
      default: asm volatile("s_wait_asynccnt 0" ::: "memory"); break;
    }
    const float4 vi = bufI[s][t];
    const float4 vt = bufT[s][t];
    mi = fmaxf(mi, fmaxf(fmaxf(vi.x, vi.y), fmaxf(vi.z, vi.w)));
    const int e0 = blkin * EPB + s * 1024 + t * 4;   // plane-local flat index
    if (vt.x > mt) { mt = vt.x; it = e0;     }
    if (vt.y > mt) { mt = vt.y; it = e0 + 1; }
    if (vt.z > mt) { mt = vt.z; it = e0 + 2; }
    if (vt.w > mt) { mt = vt.w; it = e0 + 3; }
  }
  // wave32 reduce (max + argmax, first-occurrence tie-break)
#pragma unroll
  for (int off = 16; off > 0; off >>= 1) {
    const float omi = __shfl_down(mi, off, 32);
    const float omt = __shfl_down(mt, off, 32);
    const int   oit = __shfl_down(it, off, 32);
    mi = fmaxf(mi, omi);
    if (omt > mt || (omt == mt && oit < it)) { mt = omt; it = oit; }
  }
  const int w = t >> 5;
  if ((t & 31) == 0) { smi[w] = mi; smt[w] = mt; sit[w] = it; }
  __syncthreads();
  if (t == 0) {
#pragma unroll
    for (int k = 1; k < 8; ++k) {
      mi = fmaxf(mi, smi[k]);
      if (smt[k] > mt || (smt[k] == mt && sit[k] < it)) { mt = smt[k]; it = sit[k]; }
    }
    bmax[0] = mi; bmax[1] = mt; bidx[0] = it;
  }
  __syncthreads();
  const float MI = bmax[0];
  const float MT = bmax[1];

  // ---- Phase B: 7 softmax sums (block-shifted, all exps <= 1), from LDS ----
  float Zi = 0.f, Zt = 0.f, Sx = 0.f, Sy = 0.f, Sii = 0.f, Stt = 0.f, Sit = 0.f;
  const float iw  = 1.0f / 512.0f;
  const float xw0 = (float)(((t * 4) & 511) + 1) * iw;   // stage-invariant
  const int ybase = blkin * 8 + (t >> 7);
#pragma unroll
  for (int s = 0; s < 4; ++s) {
    const float4 vi = bufI[s][t];
    const float4 vt = bufT[s][t];
    const float e0 = __expf(vi.x - MI), e1 = __expf(vi.y - MI),
                e2 = __expf(vi.z - MI), e3 = __expf(vi.w - MI);
    const float f0 = __expf(vt.x - MT), f1 = __expf(vt.y - MT),
                f2 = __expf(vt.z - MT), f3 = __expf(vt.w - MT);
    const float es = (e0 + e1) + (e2 + e3);
    const float yw = (float)(ybase + s * 2 + 1) * iw;
    Zi  += es;
    Zt  += (f0 + f1) + (f2 + f3);
    Sx  += e0 * xw0 + e1 * (xw0 + iw) + e2 * (xw0 + 2.f * iw) + e3 * (xw0 + 3.f * iw);
    Sy  += es * yw;
    Sii += (e0 * e0 + e1 * e1) + (e2 * e2 + e3 * e3);
    Stt += (f0 * f0 + f1 * f1) + (f2 * f2 + f3 * f3);
    Sit += (e0 * f0 + e1 * f1) + (e2 * f2 + e3 * f3);
  }
  float v[7] = {Zi, Zt, Sx, Sy, Sii, Stt, Sit};
#pragma unroll
  for (int off = 16; off > 0; off >>= 1) {
#pragma unroll
    for (int k = 0; k < 7; ++k) v[k] += __shfl_down(v[k], off, 32);
  }
  if ((t & 31) == 0) {
#pragma unroll
    for (int k = 0; k < 7; ++k) red[w][k] = v[k];
  }
  __syncthreads();
  if (t == 0) {
    float* row = ws + OFF_PART + blk * 16;
    row[0] = MI;
    row[1] = MT;
    reinterpret_cast<int*>(row)[2] = bidx[0];
#pragma unroll
    for (int k = 0; k < 7; ++k) {
      float a = red[0][k];
#pragma unroll
      for (int j = 1; j < 8; ++j) a += red[j][k];
      row[3 + k] = a;
    }
  }
}

// ------------------------------------------------- per-plane online merge ---
__device__ __forceinline__ void merge_partial(float& mi, float& mt, int& it, float* s,
                                              float mi2, float mt2, int it2,
                                              const float* s2) {
  const float MI = fmaxf(mi, mi2), MT = fmaxf(mt, mt2);
  const float fa = __expf(mi - MI),  fb = __expf(mi2 - MI);
  const float ga = __expf(mt - MT),  gb = __expf(mt2 - MT);
  if (mt2 > mt || (mt2 == mt && it2 < it)) it = it2;
  s[0] = s[0] * fa        + s2[0] * fb;        // Zi
  s[1] = s[1] * ga        + s2[1] * gb;        // Zt
  s[2] = s[2] * fa        + s2[2] * fb;        // Sx
  s[3] = s[3] * fa        + s2[3] * fb;        // Sy
  s[4] = s[4] * fa * fa   + s2[4] * fb * fb;   // Sii
  s[5] = s[5] * ga * ga   + s2[5] * gb * gb;   // Stt
  s[6] = s[6] * fa * ga   + s2[6] * fb * gb;   // Sit
  mi = MI; mt = MT;
}

__global__ __launch_bounds__(32) void k_comb(float* __restrict__ ws) {
  const int plane = blockIdx.x;
  const int j     = threadIdx.x;
  const float* ra = ws + OFF_PART + (plane * BPP + j) * 16;
  const float* rb = ws + OFF_PART + (plane * BPP + j + 32) * 16;

  float mi = ra[0], mt = ra[1];
  int   it = reinterpret_cast<const int*>(ra)[2];
  float s[7];
#pragma unroll
  for (int k = 0; k < 7; ++k) s[k] = ra[3 + k];
  {
    float s2[7];
#pragma unroll
    for (int k = 0; k < 7; ++k) s2[k] = rb[3 + k];
    merge_partial(mi, mt, it, s,
                  rb[0], rb[1], reinterpret_cast<const int*>(rb)[2], s2);
  }
#pragma unroll
  for (int off = 16; off > 0; off >>= 1) {
    const float omi = __shfl_down(mi, off, 32);
    const float omt = __shfl_down(mt, off, 32);
    const int   oit = __shfl_down(it, off, 32);
    float s2[7];
#pragma unroll
    for (int k = 0; k < 7; ++k) s2[k] = __shfl_down(s[k], off, 32);
    merge_partial(mi, mt, it, s, omi, omt, oit, s2);
  }
  if (j == 0) {
    const float Zi = s[0], Zt = s[1];
    const float px = s[2] / Zi;
    const float py = s[3] / Zi;
    const float mse = (s[4] / (Zi * Zi) - 2.0f * s[6] / (Zi * Zt) + s[5] / (Zt * Zt))
                      * (1.0f / (float)HW_CONST);
    const float tx = (float)((it & 511) + 1) * (1.0f / 512.0f);
    const float ty = (float)((it >> 9) + 1) * (1.0f / 512.0f);
    const float dx = tx - px, dy = ty - py;
    float* o = ws + OFF_S3 + plane * 8;
    o[0] = px; o[1] = py; o[2] = mse;
    o[3] = sqrtf(dx * dx + dy * dy);
    o[4] = tx; o[5] = ty;
  }
}

// ----------------------------------------------------------- final reduce ---
__global__ __launch_bounds__(32) void k_final(const float* __restrict__ ws,
                                              float* __restrict__ out) {
  const int b = threadIdx.x;                    // 0..31 = batch index
  const float* r0 = ws + OFF_S3 + (b * 2) * 8;
  const float* r1 = ws + OFF_S3 + (b * 2 + 1) * 8;
  float acc = r0[2] + r0[3] + r1[2] + r1[3];    // mse+ed over both channels
  const float dpx = r0[0] - r1[0], dpy = r0[1] - r1[1];
  const float dtx = r0[4] - r1[4], dty = r0[5] - r1[5];
  const float pd = sqrtf(dpx * dpx + dpy * dpy);
  const float td = sqrtf(dtx * dtx + dty * dty);
  acc += fabsf(pd - td);
#pragma unroll
  for (int off = 16; off > 0; off >>= 1) acc += __shfl_down(acc, off, 32);
  if (b == 0) out[0] = acc * (1.0f / 32.0f);
}

extern "C" void kernel_launch(void* const* d_in, const int* in_sizes, int n_in,
                              void* d_out, int out_size, void* d_ws, size_t ws_size,
                              hipStream_t stream) {
  const float* in = (const float*)d_in[0];
  const float* tg = (const float*)d_in[1];
  float* ws  = (float*)d_ws;   // needs ~258 KB
  float* out = (float*)d_out;

  k_main <<<NBLK, 256, 0, stream>>>(in, tg, ws);
  k_comb <<<  64,  32, 0, stream>>>(ws);
  k_final<<<   1,  32, 0, stream>>>(ws, out);
}